// PoincareBatchNorm_21199958573133
// MI455X (gfx1250) — compile-verified
//
#include <hip/hip_runtime.h>
#include <math.h>

// PoincareBatchNorm forward, MI455X (gfx1250, wave32).
// B=64, N=1024, D=256, f32. Memory-bound: 3 streaming passes over x (64MB,
// fits in 192MB L2) + 64MB store. Final pass uses V_WMMA_F32_16X16X4_F32 for
// the rank-2 term P*beta + Q*mean per 16-row tile; x loads / out stores in
// that pass are non-temporal (last use / dying stream).

#define B_DIM 64
#define N_DIM 1024
#define D_DIM 256
#define NPTS  (B_DIM * N_DIM)     // 65536 points
#define EPS   1e-5f
#define FMIN  1e-15f

typedef __attribute__((ext_vector_type(2))) float v2f;
typedef __attribute__((ext_vector_type(8))) float v8f;

// workspace layout (float offsets)
constexpr int OFF_Y2   = 0;                 // [NPTS]  |x|^2 per point
constexpr int OFF_MY   = OFF_Y2 + NPTS;     // [NPTS]  <mean,x>
constexpr int OFF_BX   = OFF_MY + NPTS;     // [NPTS]  <beta,x>
constexpr int OFF_NOM  = OFF_BX + NPTS;     // [B*D]   sum_n lam*x
constexpr int OFF_DEN  = OFF_NOM + B_DIM*D_DIM; // [B] sum_n (lam-1)
constexpr int OFF_MEAN = OFF_DEN + B_DIM;   // [D]     final mean
constexpr int OFF_SCAL = OFF_MEAN + D_DIM;  // [8]     m2, bm, beta2, var
constexpr int OFF_VPAR = OFF_SCAL + 8;      // [256]   var partials

__device__ __forceinline__ float wave_sum32(float v) {
#pragma unroll
  for (int m = 16; m >= 1; m >>= 1) v += __shfl_xor(v, m, 32);
  return v;
}

__device__ __forceinline__ float block_sum256(float v, float* red) {
  const int t = threadIdx.x;
  red[t] = v;
  __syncthreads();
#pragma unroll
  for (int s = 128; s >= 1; s >>= 1) {
    if (t < s) red[t] += red[t + s];
    __syncthreads();
  }
  float r = red[0];
  __syncthreads();
  return r;
}

// ---------------- Pass A: |x|^2 per point + per-batch midpoint sums --------
__global__ void __launch_bounds__(256) k_pass_a(const float* __restrict__ x,
                                                float* __restrict__ ws) {
  const int b    = blockIdx.x;         // one batch per block
  const int tid  = threadIdx.x;
  const int wave = tid >> 5;
  const int lane = tid & 31;
  const float* xb = x + (size_t)b * N_DIM * D_DIM;

  float acc0=0.f,acc1=0.f,acc2=0.f,acc3=0.f,acc4=0.f,acc5=0.f,acc6=0.f,acc7=0.f;
  float densum = 0.f;

  for (int n = wave; n < N_DIM; n += 8) {
    const float* row = xb + (size_t)n * D_DIM + lane * 8;
    __builtin_prefetch(row + 8 * D_DIM, 0, 1);       // global_prefetch_b8
    float4 a = *(const float4*)(row);
    float4 c = *(const float4*)(row + 4);
    float p = a.x*a.x + a.y*a.y + a.z*a.z + a.w*a.w
            + c.x*c.x + c.y*c.y + c.z*c.z + c.w*c.w;
    float y2 = wave_sum32(p);
    float lam = 2.0f / fmaxf(1.0f - y2, FMIN);
    if (lane == 0) ws[OFF_Y2 + b * N_DIM + n] = y2;
    acc0 += lam * a.x; acc1 += lam * a.y; acc2 += lam * a.z; acc3 += lam * a.w;
    acc4 += lam * c.x; acc5 += lam * c.y; acc6 += lam * c.z; acc7 += lam * c.w;
    densum += (lam - 1.0f);
  }

  __shared__ float s_acc[8 * 256];
  __shared__ float s_den[8];
  const int base = wave * 256 + lane * 8;
  s_acc[base + 0] = acc0; s_acc[base + 1] = acc1;
  s_acc[base + 2] = acc2; s_acc[base + 3] = acc3;
  s_acc[base + 4] = acc4; s_acc[base + 5] = acc5;
  s_acc[base + 6] = acc6; s_acc[base + 7] = acc7;
  if (lane == 0) s_den[wave] = densum;
  __syncthreads();

  float nom = 0.f;
#pragma unroll
  for (int w = 0; w < 8; ++w) nom += s_acc[w * 256 + tid];
  ws[OFF_NOM + b * D_DIM + tid] = nom;
  if (tid == 0) {
    float d = 0.f;
#pragma unroll
    for (int w = 0; w < 8; ++w) d += s_den[w];
    ws[OFF_DEN + b] = d;
  }
}

// ---------------- Midpoints (axis=1 finish, axis=0) + scalars --------------
__global__ void __launch_bounds__(256) k_mid(const float* __restrict__ beta,
                                             float* __restrict__ ws) {
  __shared__ float red[256];
  const int d = threadIdx.x;

  float nom2 = 0.f, den2 = 0.f;
  for (int b = 0; b < B_DIM; ++b) {
    float nom  = ws[OFF_NOM + b * D_DIM + d];
    float denb = fmaxf(ws[OFF_DEN + b], FMIN);
    float tm   = nom / denb;                       // two_mean component
    float t2   = block_sum256(tm * tm, red);       // |two_mean|^2 (uniform)
    float root = sqrtf(fmaxf(1.f - t2, 0.f));
    float mb   = tm / (1.f + root);                // midpoint of batch b
    float mb2  = t2 / ((1.f + root) * (1.f + root));
    float lam  = 2.f / fmaxf(1.f - mb2, FMIN);
    nom2 += lam * mb;
    den2 += (lam - 1.f);                           // uniform across threads
  }
  den2 = fmaxf(den2, FMIN);
  float tm2  = nom2 / den2;
  float t22  = block_sum256(tm2 * tm2, red);
  float root = sqrtf(fmaxf(1.f - t22, 0.f));
  float mean_d = tm2 / (1.f + root);
  float m2   = t22 / ((1.f + root) * (1.f + root));
  ws[OFF_MEAN + d] = mean_d;

  float bd    = beta[d];
  float beta2 = block_sum256(bd * bd, red);
  float bm    = block_sum256(bd * mean_d, red);
  if (d == 0) {
    ws[OFF_SCAL + 0] = m2;
    ws[OFF_SCAL + 1] = bm;
    ws[OFF_SCAL + 2] = beta2;
  }
}

// ---------------- Pass B: <mean,x>, <beta,x>, var partials -----------------
__global__ void __launch_bounds__(256) k_pass_b(const float* __restrict__ x,
                                                const float* __restrict__ beta,
                                                float* __restrict__ ws) {
  const int tid  = threadIdx.x;
  const int wave = tid >> 5;
  const int lane = tid & 31;

  float mr[8], br[8];
#pragma unroll
  for (int j = 0; j < 8; ++j) {
    mr[j] = ws[OFF_MEAN + lane * 8 + j];
    br[j] = beta[lane * 8 + j];
  }
  const float m2 = ws[OFF_SCAL + 0];
  const float Bc = 1.f - m2;

  float vacc = 0.f;
  const int r0 = (blockIdx.x * 8 + wave) * 32;    // 32 rows per wave
  for (int i = 0; i < 32; ++i) {
    const int r = r0 + i;
    const float* row = x + (size_t)r * D_DIM + lane * 8;
    __builtin_prefetch(row + D_DIM, 0, 1);
    float4 a = *(const float4*)(row);
    float4 c = *(const float4*)(row + 4);
    float pmy = mr[0]*a.x + mr[1]*a.y + mr[2]*a.z + mr[3]*a.w
              + mr[4]*c.x + mr[5]*c.y + mr[6]*c.z + mr[7]*c.w;
    float pbx = br[0]*a.x + br[1]*a.y + br[2]*a.z + br[3]*a.w
              + br[4]*c.x + br[5]*c.y + br[6]*c.z + br[7]*c.w;
    float my = wave_sum32(pmy);
    float bx = wave_sum32(pbx);
    float y2 = ws[OFF_Y2 + r];
    if (lane == 0) { ws[OFF_MY + r] = my; ws[OFF_BX + r] = bx; }
    float A_  = 1.f - 2.f*my + y2;
    float den = fmaxf(1.f - 2.f*my + m2*y2, FMIN);
    float num2 = A_*A_*m2 - 2.f*A_*Bc*my + Bc*Bc*y2;
    float sn  = sqrtf(fmaxf(num2 / (den*den), FMIN));
    float snc = fminf(sn, 1.f - 1e-7f);
    vacc += 2.f * atanhf(snc);                    // 2*||x_T|| (lambda cancels)
  }

  __shared__ float s_v[8];
  if (lane == 0) s_v[wave] = vacc;
  __syncthreads();
  if (tid == 0) {
    float s = 0.f;
#pragma unroll
    for (int w = 0; w < 8; ++w) s += s_v[w];
    ws[OFF_VPAR + blockIdx.x] = s;
  }
}

__global__ void __launch_bounds__(256) k_var(float* __restrict__ ws) {
  __shared__ float red[256];
  float v = ws[OFF_VPAR + threadIdx.x];
  float s = block_sum256(v, red);
  if (threadIdx.x == 0) ws[OFF_SCAL + 3] = s / (float)NPTS;
}

// ---------------- Pass C: out = P*beta + Q*mean + R*x (WMMA rank-2) --------
__global__ void __launch_bounds__(256) k_pass_c(const float* __restrict__ x,
                                                const float* __restrict__ beta,
                                                const float* __restrict__ gamma,
                                                const float* __restrict__ ws,
                                                float* __restrict__ out) {
  const int lane = threadIdx.x & 31;
  const int wave = threadIdx.x >> 5;
  const int l15  = lane & 15;

  const float m2    = ws[OFF_SCAL + 0];
  const float bm    = ws[OFF_SCAL + 1];
  const float beta2 = ws[OFF_SCAL + 2];
  const float var   = ws[OFF_SCAL + 3];
  const float gam   = gamma[0];
  const float Bc    = 1.f - m2;
  const float lam_b = 2.f / fmaxf(1.f - beta2, FMIN);
  const float kc    = 0.5f * gam / (var + EPS) * (1.f - beta2);

  // B fragments: K0 row = beta chunk, K1 row = mean chunk, K2/K3 = 0
  float betf[16], mef[16];
#pragma unroll
  for (int ch = 0; ch < 16; ++ch) {
    betf[ch] = (lane < 16) ? beta[ch * 16 + l15] : 0.f;
    mef[ch]  = (lane < 16) ? ws[OFF_MEAN + ch * 16 + l15] : 0.f;
  }

  for (int it = 0; it < 2; ++it) {
    const int t    = (blockIdx.x * 8 + wave) * 2 + it;   // 4096 tiles total
    const int row0 = t * 16;
    const int r    = row0 + l15;

    float y2 = ws[OFF_Y2 + r];
    float my = ws[OFF_MY + r];
    float bx = ws[OFF_BX + r];

    float A_   = 1.f - 2.f*my + y2;
    float den  = fmaxf(1.f - 2.f*my + m2*y2, FMIN);
    float num2 = A_*A_*m2 - 2.f*A_*Bc*my + Bc*Bc*y2;
    float sn   = sqrtf(fmaxf(num2 / (den*den), FMIN));
    float snc  = fminf(sn, 1.f - 1e-7f);
    float arth = atanhf(snc);
    float cc   = (arth / sn) * kc;                 // x_T = cc * sub
    float un   = sqrtf(fmaxf(cc*cc*sn*sn, FMIN));
    float tt   = tanhf(0.5f * lam_b * un);
    float tu   = tt / un;
    float bu   = cc * (Bc*bx - A_*bm) / den;       // <beta, u>
    float bs   = tu * bu;
    float s2   = tt * tt;
    float mden = fmaxf(1.f + 2.f*bs + beta2*s2, FMIN);
    float P    = (1.f + 2.f*bs + s2) / mden;
    float S    = (1.f - beta2) / mden;
    float Q    = S * (-(tu * cc * A_ / den));
    float R    = S * (tu * cc * Bc / den);

    // A fragment: 16x4, rows = points, K = [P, Q, 0, 0]
    v2f Aop;
    Aop.x = (lane < 16) ? P : 0.f;
    Aop.y = (lane < 16) ? Q : 0.f;

    float Rg[8];
#pragma unroll
    for (int g = 0; g < 8; ++g)
      Rg[g] = __shfl(R, g + ((lane & 16) >> 1), 32);

#pragma unroll
    for (int ch = 0; ch < 16; ++ch) {            // full unroll: static betf/mef
      v2f Bop;
      Bop.x = betf[ch];
      Bop.y = mef[ch];
      v8f Cz = {};
      v8f Ct = __builtin_amdgcn_wmma_f32_16x16x4_f32(
          false, Aop, false, Bop, (short)0, Cz, false, false);
      const int dbase = ch * 16 + l15;
#pragma unroll
      for (int g = 0; g < 8; ++g) {
        const int m = g + ((lane & 16) >> 1);
        const size_t idx = (size_t)(row0 + m) * D_DIM + dbase;
        float xv = __builtin_nontemporal_load(&x[idx]);     // last read of x
        __builtin_nontemporal_store(Ct[g] + Rg[g] * xv, &out[idx]);
      }
    }
  }
}

extern "C" void kernel_launch(void* const* d_in, const int* in_sizes, int n_in,
                              void* d_out, int out_size, void* d_ws, size_t ws_size,
                              hipStream_t stream) {
  (void)in_sizes; (void)n_in; (void)out_size; (void)ws_size;
  const float* x     = (const float*)d_in[0];
  const float* beta  = (const float*)d_in[1];
  const float* gamma = (const float*)d_in[2];
  float* out = (float*)d_out;
  float* ws  = (float*)d_ws;

  k_pass_a<<<B_DIM, 256, 0, stream>>>(x, ws);
  k_mid   <<<1,     256, 0, stream>>>(beta, ws);
  k_pass_b<<<256,   256, 0, stream>>>(x, beta, ws);
  k_var   <<<1,     256, 0, stream>>>(ws);
  k_pass_c<<<256,   256, 0, stream>>>(x, beta, gamma, ws, out);
}